// NodeKalmanGnnRnn_38173669327370
// MI455X (gfx1250) — compile-verified
//
#include <hip/hip_runtime.h>
#include <hip/hip_bf16.h>

typedef __attribute__((ext_vector_type(2))) float v2f;
typedef __attribute__((ext_vector_type(8))) float v8f;

// ---------------------------------------------------------------------------
// WMMA fragment helpers (V_WMMA_F32_16X16X4_F32, wave32)
//
// A (16x4 f32): lane 0-15  -> row M=lane,     v0=K0, v1=K1
//               lane 16-31 -> row M=lane-16,  v0=K2, v1=K3
// B (4x16 f32): lane 0-15  -> col N=lane,     v0=K0, v1=K1
//               lane 16-31 -> col N=lane-16,  v0=K2, v1=K3
// C/D (16x16 f32, v8f): element j -> row M = j + 8*(lane>=16), col N = lane&15
// ---------------------------------------------------------------------------

static __device__ __forceinline__ v8f frag_splat(float v) {
    v8f r;
#pragma unroll
    for (int j = 0; j < 8; ++j) r[j] = v;
    return r;
}

// Bias broadcast: col = lane&15 for every element of the fragment.
static __device__ __forceinline__ v8f bias_frag(const float* b) {
    return frag_splat(b[threadIdx.x & 15]);
}

// Elementwise load of a 16x16 tile in C/D layout from a row-major matrix.
static __device__ __forceinline__ v8f load_frag(const float* src, int ld) {
    int lane = threadIdx.x & 31;
    int lh = lane >> 4, li = lane & 15;
    v8f r;
#pragma unroll
    for (int j = 0; j < 8; ++j) r[j] = src[(j + 8 * lh) * (long)ld + li];
    return r;
}

static __device__ __forceinline__ void store_frag(float* dst, int ld, v8f a, bool leaky) {
    int lane = threadIdx.x & 31;
    int lh = lane >> 4, li = lane & 15;
#pragma unroll
    for (int j = 0; j < 8; ++j) {
        float v = a[j];
        if (leaky) v = (v > 0.0f) ? v : 0.01f * v;
        dst[(j + 8 * lh) * (long)ld + li] = v;
    }
}

// D += X(16xK, row-major, ldx) @ W(M,K row-major; pass pointer at row mBase)^T
// K must be a multiple of 4.
static __device__ __forceinline__ v8f gemm16(const float* __restrict__ X, int ldx,
                                             const float* __restrict__ W, int ldw,
                                             int K, v8f acc) {
    int lane = threadIdx.x & 31;
    int lh = lane >> 4, li = lane & 15;
    const float* xr = X + (long)li * ldx;   // node row for this lane
    const float* wr = W + (long)li * ldw;   // weight row (output col) for this lane
    for (int k = 0; k < K; k += 4) {
        v2f a, b;
        a.x = xr[k + 2 * lh + 0];
        a.y = xr[k + 2 * lh + 1];
        b.x = wr[k + 2 * lh + 0];
        b.y = wr[k + 2 * lh + 1];
        acc = __builtin_amdgcn_wmma_f32_16x16x4_f32(false, a, false, b,
                                                    (short)0, acc, false, false);
    }
    return acc;
}

static __device__ __forceinline__ float sigm(float x) { return 1.0f / (1.0f + expf(-x)); }

// ---------------------------------------------------------------------------
// Kernel A: dy-MLP, r-MLP, GRU_r, fused GCN projection xw = [dy|h_r] @ gcn_w^T
// 2 waves per block, one 16-node tile per wave.
// ---------------------------------------------------------------------------
__global__ void __launch_bounds__(64) k_node_a(
    const float* __restrict__ yinn,   // [N,72]
    const float* __restrict__ dyi,    // [N,8]
    const float* __restrict__ hidden, // [N,64]
    const float* __restrict__ w_dy1, const float* __restrict__ b_dy1,
    const float* __restrict__ w_dy2, const float* __restrict__ b_dy2,
    const float* __restrict__ w_r1,  const float* __restrict__ b_r1,
    const float* __restrict__ w_r2,  const float* __restrict__ b_r2,
    const float* __restrict__ wih,   const float* __restrict__ whh,
    const float* __restrict__ bih,   const float* __restrict__ bhh,
    const float* __restrict__ gcn_w,
    float* __restrict__ out_hr,      // [N,64]
    float* __restrict__ xw,          // [N,64] workspace
    int N) {
    __shared__ float s_t[2][16 * 64];
    __shared__ float s_rg[2][16 * 64];
    __shared__ float s_g[2][16 * 128];

    int wave = threadIdx.x >> 5;
    int node0 = (blockIdx.x * 2 + wave) * 16;
    if (node0 >= N) return;
    float* st = s_t[wave];
    float* srg = s_rg[wave];
    float* sg = s_g[wave];

    // --- dy = leaky(leaky(y72 @ w_dy1^T + b) @ w_dy2^T + b) -> sg[:,0:64]
    for (int m = 0; m < 4; ++m) {
        v8f a = bias_frag(b_dy1 + m * 16);
        a = gemm16(yinn + (long)node0 * 72, 72, w_dy1 + m * 16 * 72, 72, 72, a);
        store_frag(st + m * 16, 64, a, true);
    }
    for (int m = 0; m < 4; ++m) {
        v8f a = bias_frag(b_dy2 + m * 16);
        a = gemm16(st, 64, w_dy2 + m * 16 * 64, 64, 64, a);
        store_frag(sg + m * 16, 128, a, true);
    }

    // --- rg = leaky(leaky(dy8 @ w_r1^T + b) @ w_r2^T + b) -> srg
    for (int m = 0; m < 4; ++m) {
        v8f a = bias_frag(b_r1 + m * 16);
        a = gemm16(dyi + (long)node0 * 8, 8, w_r1 + m * 16 * 8, 8, 8, a);
        store_frag(st + m * 16, 64, a, true);
    }
    for (int m = 0; m < 4; ++m) {
        v8f a = bias_frag(b_r2 + m * 16);
        a = gemm16(st, 64, w_r2 + m * 16 * 64, 64, 64, a);
        store_frag(srg + m * 16, 64, a, true);
    }

    // --- GRU_r: h_new = (1-z)*n + z*h, per 16-col group -> sg[:,64:128] + out_hr
    const float* hbase = hidden + (long)node0 * 64;
    for (int g = 0; g < 4; ++g) {
        v8f ir = bias_frag(bih + g * 16);
        ir = gemm16(srg, 64, wih + (g * 16) * 64, 64, 64, ir);
        v8f iz = bias_frag(bih + 64 + g * 16);
        iz = gemm16(srg, 64, wih + (64 + g * 16) * 64, 64, 64, iz);
        v8f in = bias_frag(bih + 128 + g * 16);
        in = gemm16(srg, 64, wih + (128 + g * 16) * 64, 64, 64, in);
        v8f hr = bias_frag(bhh + g * 16);
        hr = gemm16(hbase, 64, whh + (g * 16) * 64, 64, 64, hr);
        v8f hz = bias_frag(bhh + 64 + g * 16);
        hz = gemm16(hbase, 64, whh + (64 + g * 16) * 64, 64, 64, hz);
        v8f hn = bias_frag(bhh + 128 + g * 16);
        hn = gemm16(hbase, 64, whh + (128 + g * 16) * 64, 64, 64, hn);
        v8f h = load_frag(hbase + g * 16, 64);
        v8f hnew;
#pragma unroll
        for (int j = 0; j < 8; ++j) {
            float r = sigm(ir[j] + hr[j]);
            float z = sigm(iz[j] + hz[j]);
            float n = tanhf(in[j] + r * hn[j]);
            hnew[j] = (1.0f - z) * n + z * h[j];
        }
        store_frag(sg + 64 + g * 16, 128, hnew, false);
        store_frag(out_hr + (long)node0 * 64 + g * 16, 64, hnew, false);
    }

    // --- xw = [dy | h_r] @ gcn_w^T   (bias deferred to kernel C)
    for (int m = 0; m < 4; ++m) {
        v8f a = frag_splat(0.0f);
        a = gemm16(sg, 128, gcn_w + m * 16 * 128, 128, 128, a);
        store_frag(xw + (long)node0 * 64 + m * 16, 64, a, false);
    }
}

// ---------------------------------------------------------------------------
// GCN edge phase
// ---------------------------------------------------------------------------
__global__ void k_deg_init(float* __restrict__ deg, int N) {
    int i = blockIdx.x * blockDim.x + threadIdx.x;
    if (i < N) deg[i] = 1.0f;  // self loop
}

__global__ void k_deg_count(const int* __restrict__ ei, float* __restrict__ deg, int E) {
    int e = blockIdx.x * blockDim.x + threadIdx.x;
    if (e < E) atomicAdd(&deg[ei[(long)E + e]], 1.0f);
}

__global__ void k_self(const float* __restrict__ xw, const float* __restrict__ deg,
                       float* __restrict__ dis, float* __restrict__ agg, int N) {
    long tid = (long)blockIdx.x * blockDim.x + threadIdx.x;
    if (tid >= (long)N * 64) return;
    int n = (int)(tid >> 6);
    float di = rsqrtf(deg[n]);  // deg >= 1 always (self loop)
    if ((tid & 63) == 0) dis[n] = di;
    agg[tid] = xw[tid] * di * di;  // self-loop message
}

__global__ void k_edges(const int* __restrict__ ei, const float* __restrict__ xw,
                        const float* __restrict__ dis, float* __restrict__ agg, int E) {
    long tid = (long)blockIdx.x * blockDim.x + threadIdx.x;
    int e = (int)(tid >> 4);
    int c0 = (int)(tid & 15);
    if (e >= E) return;
    int r = ei[e];
    int c = ei[(long)E + e];
    float nrm = dis[r] * dis[c];
    const float* src = xw + (long)r * 64;
    float* dst = agg + (long)c * 64;
#pragma unroll
    for (int j = 0; j < 4; ++j) {
        int col = c0 + 16 * j;
        atomicAdd(dst + col, src[col] * nrm);
    }
}

// ---------------------------------------------------------------------------
// Kernel C: node_feat = agg/deg + b, output head, signal MLP, sigma GRU
// ---------------------------------------------------------------------------
__global__ void __launch_bounds__(64) k_node_c(
    const float* __restrict__ psig,  // [N,16]
    const float* __restrict__ dxf,   // [N,16]
    const float* __restrict__ agg,   // [N,64]
    const float* __restrict__ deg,   // [N]
    const float* __restrict__ gcn_b,
    const float* __restrict__ w_o1, const float* __restrict__ b_o1,
    const float* __restrict__ w_o2, const float* __restrict__ b_o2,
    const float* __restrict__ w_o3, const float* __restrict__ b_o3,
    const float* __restrict__ w_s1, const float* __restrict__ b_s1,
    const float* __restrict__ w_s2, const float* __restrict__ b_s2,
    const float* __restrict__ swih, const float* __restrict__ swhh,
    const float* __restrict__ sbih, const float* __restrict__ sbhh,
    float* __restrict__ out_nk,   // [N,16]
    float* __restrict__ out_sig,  // [N,16]
    int N) {
    __shared__ float s_nf[2][16 * 80];
    __shared__ float s_h1[2][16 * 128];
    __shared__ float s_h2[2][16 * 64];
    __shared__ float s_nk[2][16 * 16];

    int wave = threadIdx.x >> 5;
    int lane = threadIdx.x & 31;
    int node0 = (blockIdx.x * 2 + wave) * 16;
    if (node0 >= N) return;
    float* nf = s_nf[wave];
    float* h1 = s_h1[wave];
    float* h2 = s_h2[wave];
    float* nk = s_nk[wave];

    // nk_in tile: cols 0..63 = agg/deg + gcn_b, cols 64..79 = pred_sigma
    for (int idx = lane; idx < 16 * 80; idx += 32) {
        int row = idx / 80, c = idx - row * 80;
        float v;
        if (c < 64)
            v = agg[(long)(node0 + row) * 64 + c] / deg[node0 + row] + gcn_b[c];
        else
            v = psig[(long)(node0 + row) * 16 + (c - 64)];
        nf[idx] = v;
    }

    // h1 = leaky(nk_in @ w_o1^T + b)  [16,128]
    for (int m = 0; m < 8; ++m) {
        v8f a = bias_frag(b_o1 + m * 16);
        a = gemm16(nf, 80, w_o1 + m * 16 * 80, 80, 80, a);
        store_frag(h1 + m * 16, 128, a, true);
    }
    // h2 = leaky(h1 @ w_o2^T + b)  [16,64]
    for (int m = 0; m < 4; ++m) {
        v8f a = bias_frag(b_o2 + m * 16);
        a = gemm16(h1, 128, w_o2 + m * 16 * 128, 128, 128, a);
        store_frag(h2 + m * 16, 64, a, true);
    }
    // nk_out = h2 @ w_o3^T + b  [16,16] — exactly one column group
    v8f nka = bias_frag(b_o3);
    nka = gemm16(h2, 64, w_o3, 64, 64, nka);
    store_frag(out_nk + (long)node0 * 16, 16, nka, false);
    store_frag(nk, 16, nka, false);

    // dx = leaky(leaky(dxf @ w_s1^T + b) @ w_s2^T + b)  (reuse h2, then h1 as 64-wide)
    for (int m = 0; m < 4; ++m) {
        v8f a = bias_frag(b_s1 + m * 16);
        a = gemm16(dxf + (long)node0 * 16, 16, w_s1 + m * 16 * 16, 16, 16, a);
        store_frag(h2 + m * 16, 64, a, true);
    }
    for (int m = 0; m < 4; ++m) {
        v8f a = bias_frag(b_s2 + m * 16);
        a = gemm16(h2, 64, w_s2 + m * 16 * 64, 64, 64, a);
        store_frag(h1 + m * 16, 64, a, true);
    }

    // sigma GRU: hidden state is nk_out (one fragment)
    v8f ir = bias_frag(sbih);
    ir = gemm16(h1, 64, swih, 64, 64, ir);
    v8f iz = bias_frag(sbih + 16);
    iz = gemm16(h1, 64, swih + 16 * 64, 64, 64, iz);
    v8f in = bias_frag(sbih + 32);
    in = gemm16(h1, 64, swih + 32 * 64, 64, 64, in);
    v8f hr = bias_frag(sbhh);
    hr = gemm16(nk, 16, swhh, 16, 16, hr);
    v8f hz = bias_frag(sbhh + 16);
    hz = gemm16(nk, 16, swhh + 16 * 16, 16, 16, hz);
    v8f hn = bias_frag(sbhh + 32);
    hn = gemm16(nk, 16, swhh + 32 * 16, 16, 16, hn);

    v8f sig;
#pragma unroll
    for (int j = 0; j < 8; ++j) {
        float r = sigm(ir[j] + hr[j]);
        float z = sigm(iz[j] + hz[j]);
        float n = tanhf(in[j] + r * hn[j]);
        sig[j] = (1.0f - z) * n + z * nka[j];
    }
    store_frag(out_sig + (long)node0 * 16, 16, sig, false);
}

// ---------------------------------------------------------------------------
extern "C" void kernel_launch(void* const* d_in, const int* in_sizes, int n_in,
                              void* d_out, int out_size, void* d_ws, size_t ws_size,
                              hipStream_t stream) {
    const float* yinn = (const float*)d_in[0];   // [N,72]
    const float* dyi = (const float*)d_in[1];    // [N,8]
    const float* dxf = (const float*)d_in[2];    // [N,16]
    const float* psig = (const float*)d_in[3];   // [N,16]
    const float* hidden = (const float*)d_in[4]; // [1,N,64]
    const int* ei = (const int*)d_in[6];         // [2,E]
    const float* w_dy1 = (const float*)d_in[7];
    const float* b_dy1 = (const float*)d_in[8];
    const float* w_dy2 = (const float*)d_in[9];
    const float* b_dy2 = (const float*)d_in[10];
    const float* w_r1 = (const float*)d_in[11];
    const float* b_r1 = (const float*)d_in[12];
    const float* w_r2 = (const float*)d_in[13];
    const float* b_r2 = (const float*)d_in[14];
    const float* wih = (const float*)d_in[15];
    const float* whh = (const float*)d_in[16];
    const float* bih = (const float*)d_in[17];
    const float* bhh = (const float*)d_in[18];
    const float* gcn_w = (const float*)d_in[19];
    const float* gcn_b = (const float*)d_in[20];
    const float* w_s1 = (const float*)d_in[21];
    const float* b_s1 = (const float*)d_in[22];
    const float* w_s2 = (const float*)d_in[23];
    const float* b_s2 = (const float*)d_in[24];
    const float* swih = (const float*)d_in[25];
    const float* swhh = (const float*)d_in[26];
    const float* sbih = (const float*)d_in[27];
    const float* sbhh = (const float*)d_in[28];
    const float* w_o1 = (const float*)d_in[29];
    const float* b_o1 = (const float*)d_in[30];
    const float* w_o2 = (const float*)d_in[31];
    const float* b_o2 = (const float*)d_in[32];
    const float* w_o3 = (const float*)d_in[33];
    const float* b_o3 = (const float*)d_in[34];

    const int N = in_sizes[0] / 72;
    const int E = in_sizes[6] / 2;

    float* ws = (float*)d_ws;
    float* xw = ws;                            // [N,64]
    float* agg = ws + (size_t)N * 64;          // [N,64]
    float* deg = ws + (size_t)N * 128;         // [N]
    float* dis = ws + (size_t)N * 128 + N;     // [N]

    float* out = (float*)d_out;
    float* out_nk = out;                       // [N,16]
    float* out_hr = out + (size_t)N * 16;      // [N,64]
    float* out_sig = out + (size_t)N * 80;     // [N,16]
    void* out_ei = (void*)(out + (size_t)N * 96);  // [2,E] int bits

    const int tiles = (N + 15) / 16;
    const int blocksA = (tiles + 1) / 2;

    k_node_a<<<blocksA, 64, 0, stream>>>(yinn, dyi, hidden,
                                         w_dy1, b_dy1, w_dy2, b_dy2,
                                         w_r1, b_r1, w_r2, b_r2,
                                         wih, whh, bih, bhh,
                                         gcn_w, out_hr, xw, N);

    k_deg_init<<<(N + 255) / 256, 256, 0, stream>>>(deg, N);
    k_deg_count<<<(E + 255) / 256, 256, 0, stream>>>(ei, deg, E);
    k_self<<<(int)(((long)N * 64 + 255) / 256), 256, 0, stream>>>(xw, deg, dis, agg, N);
    k_edges<<<(int)(((long)E * 16 + 255) / 256), 256, 0, stream>>>(ei, xw, dis, agg, E);

    k_node_c<<<blocksA, 64, 0, stream>>>(psig, dxf, agg, deg, gcn_b,
                                         w_o1, b_o1, w_o2, b_o2, w_o3, b_o3,
                                         w_s1, b_s1, w_s2, b_s2,
                                         swih, swhh, sbih, sbhh,
                                         out_nk, out_sig, N);

    hipMemcpyAsync(out_ei, d_in[6], (size_t)2 * E * sizeof(int),
                   hipMemcpyDeviceToDevice, stream);
}